// MTH_diff_18622978195604
// MI455X (gfx1250) — compile-verified
//
#include <hip/hip_runtime.h>
#include <cstdint>

// Memory-bound temporal spiking scan (no GEMM -> WMMA inapplicable).
// Roofline: 167.8 MB @ 23.3 TB/s ~= 7.2 us. VALU trimmed below that via a
// frexp/ldexp first-crossing trick (thresholds thresh/2^k are monotone in k),
// with the frexp/ldexp shared between the pos and neg branches (frexp is
// sign-agnostic). Loads use the CDNA5 async-to-LDS path (ASYNCcnt) with a
// single s_wait_asynccnt; stores are non-temporal b128.

#ifndef USE_ASYNC_LDS
#define USE_ASYNC_LDS 1
#endif

typedef float v4f __attribute__((ext_vector_type(4)));

static constexpr int       kT     = 4;                      // timesteps
static constexpr long long kN     = 32LL * 128 * 32 * 32;   // elems per timestep
static constexpr long long kNV    = kN / 4;                 // float4 per timestep
static constexpr int       kBlock = 256;                    // 8 wave32

__device__ __forceinline__ int fexp(float v) {
#if __has_builtin(__builtin_amdgcn_frexp_expf)
  return __builtin_amdgcn_frexp_expf(v);   // v_frexp_exp_i32_f32 (exponent of |v|)
#else
  int e; (void)__builtin_frexpf(v, &e); return e;
#endif
}

// th * 2^min(0, e-1)  ==  (th/2) * 2^min(1, e)
__device__ __forceinline__ float mag_from_exp(float th_half, int e) {
  return __builtin_ldexpf(th_half, e > 1 ? 1 : e);   // v_min_i32 + v_ldexp_f32
}

__global__ __launch_bounds__(kBlock) void mth_snn_kernel(
    const float* __restrict__ x, const float* __restrict__ thresh,
    float* __restrict__ out) {
  const int tid   = threadIdx.x;
  const long long i = (long long)blockIdx.x * kBlock + tid;  // float4 index

  const float th      = thresh[0];
  const float th_half = 0.5f * th;
  const float sc_m    = 1.0f / (0.75f * th);   // pos/neg membrane threshold scale
  const float sc_c    = 1.0f / th;             // cum threshold scale
  constexpr float kVmin = 0.0078125f;          // 2^-7: smallest threshold (k=7)

  const v4f* __restrict__ xv = (const v4f*)x;
  v4f* __restrict__ ov = (v4f*)out;

  v4f bias, xt[kT];

#if USE_ASYNC_LDS
  __shared__ v4f tile[kT + 1][kBlock];      // 20 KB / block
  {
    const unsigned lds0 = (unsigned)(uintptr_t)(&tile[0][tid]);
#pragma unroll
    for (int t = 0; t <= kT; ++t) {
      const uint64_t ga = (uint64_t)(xv + (long long)t * kNV + i);
      const unsigned la = lds0 + (unsigned)(t * kBlock * (int)sizeof(v4f));
      asm volatile("global_load_async_to_lds_b128 %0, %1, off"
                   :: "v"(la), "v"(ga) : "memory");
    }
    asm volatile("s_wait_asynccnt 0x0" ::: "memory");
    bias = tile[0][tid];
#pragma unroll
    for (int t = 0; t < kT; ++t) xt[t] = tile[t + 1][tid];
  }
#else
  bias = xv[i];
#pragma unroll
  for (int t = 0; t < kT; ++t) xt[t] = xv[(long long)(t + 1) * kNV + i];
#endif

  const v4f z = {0.0f, 0.0f, 0.0f, 0.0f};
  __builtin_nontemporal_store(z, ov + i);   // output slice 0 is zeros

  v4f mem = z, cum = z, eout = z, ein = bias;

#pragma unroll
  for (int t = 0; t < kT; ++t) {
    const float rcp = 1.0f / (float)(t + 1);   // constant-folded per step
    v4f spike;
#pragma unroll
    for (int c = 0; c < 4; ++c) {
      const float d  = xt[t][c] - bias[c];
      const float m  = mem[c] + d + (ein[c] - eout[c]);
      const float cu = cum[c] + eout[c];

      // Membrane branch: vm = m / (0.75*th). frexp exponent is sign-agnostic,
      // so pos (first k with m >= 0.75*th/2^k) and negA (first k with
      // m <= -0.75*th/2^k) share one magnitude th/2^k* = (th/2)*2^min(1,e).
      const float vm   = m * sc_m;
      const float smag = mag_from_exp(th_half, fexp(vm));
      const float sp   = (vm >=  kVmin) ? smag : 0.0f;
      const float snA  = (vm <= -kVmin) ? smag : 0.0f;

      // Cum branch: first k with cu >= th/2^k.
      const float vc   = cu * sc_c;
      const float snB  = (vc >= kVmin) ? mag_from_exp(th_half, fexp(vc)) : 0.0f;

      // neg fires at k = max(kA,kB) -> thre[max] = min of the two magnitudes;
      // min with 0 correctly kills it when either condition never holds.
      const float s = sp - fminf(snA, snB);

      mem[c]  = m - s;
      cum[c]  = cu + s;
      ein[c]  = fmaf(d, rcp, ein[c]);
      eout[c] = fmaf(s, rcp, eout[c]);
      spike[c] = s;
    }
    __builtin_nontemporal_store(spike, ov + (long long)(t + 1) * kNV + i);
  }
}

extern "C" void kernel_launch(void* const* d_in, const int* in_sizes, int n_in,
                              void* d_out, int out_size, void* d_ws, size_t ws_size,
                              hipStream_t stream) {
  (void)in_sizes; (void)n_in; (void)out_size; (void)d_ws; (void)ws_size;
  const float* x      = (const float*)d_in[0];
  const float* thresh = (const float*)d_in[1];
  float* out          = (float*)d_out;
  const unsigned grid = (unsigned)(kNV / kBlock);   // 4096 blocks, exact cover
  mth_snn_kernel<<<dim3(grid), dim3(kBlock), 0, stream>>>(x, thresh, out);
}